// EuclideanAttention_78829829750904
// MI455X (gfx1250) — compile-verified
//
#include <hip/hip_runtime.h>
#include <hip/hip_bf16.h>

typedef __attribute__((ext_vector_type(2))) float v2f;
typedef __attribute__((ext_vector_type(8))) float v8f;

#define D_MODEL 1024
#define N_HEADS 16
#define D_HEAD  64
#define BATCH   4
#define SEQ     2048
#define ROWS    (BATCH * SEQ)   // 8192

static __device__ __forceinline__ v8f wmma_f32(v2f a, v2f b, v8f c) {
  // D = A(16x4,f32) * B(4x16,f32) + C(16x16,f32)
  return __builtin_amdgcn_wmma_f32_16x16x4_f32(
      /*neg_a=*/false, a, /*neg_b=*/false, b,
      /*c_mod=*/(short)0, c, /*reuse_a=*/false, /*reuse_b=*/false);
}

// ---------------------------------------------------------------------------
// Kernel 0: transpose the four 1024x1024 weight matrices (W -> W^T) so that
// WMMA B-fragments become contiguous 8-byte loads in the GEMM kernels.
// ---------------------------------------------------------------------------
__global__ __launch_bounds__(256) void transpose_w_kernel(
    const float* __restrict__ W0, const float* __restrict__ W1,
    const float* __restrict__ W2, const float* __restrict__ W3,
    float* __restrict__ T0, float* __restrict__ T1,
    float* __restrict__ T2, float* __restrict__ T3) {
  const float* W = (blockIdx.z == 0) ? W0 : (blockIdx.z == 1) ? W1
                   : (blockIdx.z == 2) ? W2 : W3;
  float* T = (blockIdx.z == 0) ? T0 : (blockIdx.z == 1) ? T1
             : (blockIdx.z == 2) ? T2 : T3;

  __shared__ float tile[32][33];
  const int tx = threadIdx.x & 31;
  const int ty = threadIdx.x >> 5;        // 0..7
  const int x0 = blockIdx.x * 32;
  const int y0 = blockIdx.y * 32;

#pragma unroll
  for (int i = 0; i < 4; ++i)
    tile[ty + i * 8][tx] = W[(size_t)(y0 + ty + i * 8) * D_MODEL + x0 + tx];
  __syncthreads();
#pragma unroll
  for (int i = 0; i < 4; ++i)
    T[(size_t)(x0 + ty + i * 8) * D_MODEL + y0 + tx] = tile[tx][ty + i * 8];
}

// ---------------------------------------------------------------------------
// Kernel 1: fused QKV projection.
//   X: (ROWS, D_MODEL) row-major;  Wt*: transposed weights (N, K) row-major.
//   Q, K written in (B*H, S, Dh); V written transposed (B*H, Dh, S).
// Each wave computes a 32x16 tile of Q, K and V (B-fragments reused 2x,
// A-fragments reused 3x; all fragment loads are contiguous b64).
// ---------------------------------------------------------------------------
__global__ __launch_bounds__(256) void qkv_proj_kernel(
    const float* __restrict__ X,
    const float* __restrict__ Wqt, const float* __restrict__ Wkt,
    const float* __restrict__ Wvt,
    float* __restrict__ Q, float* __restrict__ K, float* __restrict__ Vt) {
  const int wave = threadIdx.x >> 5;
  const int lane = threadIdx.x & 31;
  const int half = lane >> 4;
  const int l16  = lane & 15;

  const int row0 = blockIdx.x * 32;
  const int col0 = (blockIdx.y * 8 + wave) * 16;
  const int col  = col0 + l16;

  v8f cq0 = {}, ck0 = {}, cv0 = {};
  v8f cq1 = {}, ck1 = {}, cv1 = {};
  const float* xrow0 = X + (size_t)(row0 + l16) * D_MODEL;
  const float* xrow1 = xrow0 + (size_t)16 * D_MODEL;
  const float* wq = Wqt + (size_t)col * D_MODEL;
  const float* wk = Wkt + (size_t)col * D_MODEL;
  const float* wv = Wvt + (size_t)col * D_MODEL;

#pragma unroll 4
  for (int k0 = 0; k0 < D_MODEL; k0 += 4) {
    const int kr = k0 + 2 * half;
    v2f a0 = *(const v2f*)(xrow0 + kr);
    v2f a1 = *(const v2f*)(xrow1 + kr);
    v2f bq = *(const v2f*)(wq + kr);
    v2f bk = *(const v2f*)(wk + kr);
    v2f bv = *(const v2f*)(wv + kr);
    cq0 = wmma_f32(a0, bq, cq0);  cq1 = wmma_f32(a1, bq, cq1);
    ck0 = wmma_f32(a0, bk, ck0);  ck1 = wmma_f32(a1, bk, ck1);
    cv0 = wmma_f32(a0, bv, cv0);  cv1 = wmma_f32(a1, bv, cv1);
  }

  const int h = col >> 6;        // head index
  const int d = col & 63;        // dim within head
#pragma unroll
  for (int r = 0; r < 8; ++r) {
#pragma unroll
    for (int t = 0; t < 2; ++t) {
      const int row = row0 + t * 16 + r + 8 * half;
      const int b = row >> 11;             // / SEQ
      const int s = row & (SEQ - 1);
      const int bh = b * N_HEADS + h;
      const size_t o  = ((size_t)bh * SEQ + s) * D_HEAD + d;       // (BH,S,Dh)
      const size_t ov = ((size_t)bh * D_HEAD + d) * SEQ + s;       // (BH,Dh,S)
      Q[o]  = t ? cq1[r] : cq0[r];
      K[o]  = t ? ck1[r] : ck0[r];
      Vt[ov] = t ? cv1[r] : cv0[r];
    }
  }
}

// ---------------------------------------------------------------------------
// Kernel 2: causal flash attention, one (b,h) per blockIdx.y.
// Each wave owns a 16-query tile; online softmax; P staged through LDS to
// convert from C-layout to A-layout for the P@V WMMAs. V is pre-transposed
// so its B-fragments are contiguous b64 loads.
// ---------------------------------------------------------------------------
__global__ __launch_bounds__(256) void flash_attn_kernel(
    const float* __restrict__ Q, const float* __restrict__ K,
    const float* __restrict__ Vt, float* __restrict__ Oout) {
  const int wave = threadIdx.x >> 5;
  const int lane = threadIdx.x & 31;
  const int half = lane >> 4;
  const int l16  = lane & 15;

  const int bh = blockIdx.y;
  const int q0 = (blockIdx.x * 8 + wave) * 16;

  const float* Qm  = Q  + (size_t)bh * SEQ * D_HEAD;
  const float* Km  = K  + (size_t)bh * SEQ * D_HEAD;
  const float* Vtm = Vt + (size_t)bh * D_HEAD * SEQ;   // (Dh, S)

  __shared__ float lds[8][16][17];   // per-wave padded P tile
  float (*pt)[17] = lds[wave];

  // Preload Q fragments for the whole d=64 K-dimension (16 steps of 4).
  v2f qf[16];
  const float* qrow = Qm + (size_t)(q0 + l16) * D_HEAD;
#pragma unroll
  for (int kk = 0; kk < 16; ++kk)
    qf[kk] = *(const v2f*)(qrow + kk * 4 + 2 * half);

  v8f o0 = {}, o1 = {}, o2 = {}, o3 = {};
  float m8[8], l8[8];
#pragma unroll
  for (int r = 0; r < 8; ++r) { m8[r] = -__builtin_inff(); l8[r] = 0.0f; }

  // Per-lane base pointers for V^T B-fragments (4 N-tiles of 16 columns).
  const float* vb0 = Vtm + (size_t)( 0 + l16) * SEQ;
  const float* vb1 = Vtm + (size_t)(16 + l16) * SEQ;
  const float* vb2 = Vtm + (size_t)(32 + l16) * SEQ;
  const float* vb3 = Vtm + (size_t)(48 + l16) * SEQ;

  for (int t0 = 0; t0 <= q0; t0 += 16) {
    // S-tile = Q(16x64) @ K^T(64x16), 16 WMMAs
    v8f sc = {};
    const float* krow = Km + (size_t)(t0 + l16) * D_HEAD;
#pragma unroll
    for (int kk = 0; kk < 16; ++kk) {
      v2f bf = *(const v2f*)(krow + kk * 4 + 2 * half);
      sc = wmma_f32(qf[kk], bf, sc);
    }

    // Scale + causal mask (only the diagonal tile needs masking).
    const bool diag = (t0 == q0);
#pragma unroll
    for (int r = 0; r < 8; ++r) {
      float v = sc[r] * 0.125f;   // 1/sqrt(64)
      if (diag && (l16 > r + 8 * half)) v = -__builtin_inff();
      sc[r] = v;
    }

    // Online softmax: per-row (half-wave) max/sum reductions.
#pragma unroll
    for (int r = 0; r < 8; ++r) {
      float v = sc[r];
      v = fmaxf(v, __shfl_xor(v, 1, 16));
      v = fmaxf(v, __shfl_xor(v, 2, 16));
      v = fmaxf(v, __shfl_xor(v, 4, 16));
      v = fmaxf(v, __shfl_xor(v, 8, 16));
      const float mn   = fmaxf(m8[r], v);
      const float resc = __expf(m8[r] - mn);
      m8[r] = mn;
      const float p = __expf(sc[r] - mn);
      sc[r] = p;
      float s = p;
      s += __shfl_xor(s, 1, 16);
      s += __shfl_xor(s, 2, 16);
      s += __shfl_xor(s, 4, 16);
      s += __shfl_xor(s, 8, 16);
      l8[r] = l8[r] * resc + s;
      o0[r] *= resc; o1[r] *= resc; o2[r] *= resc; o3[r] *= resc;
    }

    // C-layout -> A-layout conversion via LDS (wave-private, no barrier).
#pragma unroll
    for (int r = 0; r < 8; ++r)
      pt[r + 8 * half][l16] = sc[r];
    asm volatile("s_wait_dscnt 0" ::: "memory");

    // O(16x64) += P(16x16) @ V(16x64): 4 K-steps x 4 N-tiles
#pragma unroll
    for (int kk = 0; kk < 4; ++kk) {
      v2f a;
      a.x = pt[l16][kk * 4 + 2 * half];
      a.y = pt[l16][kk * 4 + 2 * half + 1];
      const int tk = t0 + kk * 4 + 2 * half;   // key index of B row pair
      v2f b;
      b = *(const v2f*)(vb0 + tk); o0 = wmma_f32(a, b, o0);
      b = *(const v2f*)(vb1 + tk); o1 = wmma_f32(a, b, o1);
      b = *(const v2f*)(vb2 + tk); o2 = wmma_f32(a, b, o2);
      b = *(const v2f*)(vb3 + tk); o3 = wmma_f32(a, b, o3);
    }
  }

  // Normalize and write attn output in (B, S, H*Dh) layout for out-proj.
  const int b = bh >> 4;
  const int h = bh & (N_HEADS - 1);
#pragma unroll
  for (int r = 0; r < 8; ++r) {
    const float inv = 1.0f / l8[r];
    const int s = q0 + r + 8 * half;
    float* orow = Oout + (size_t)(b * SEQ + s) * (N_HEADS * D_HEAD) + h * D_HEAD;
    orow[ 0 + l16] = o0[r] * inv;
    orow[16 + l16] = o1[r] * inv;
    orow[32 + l16] = o2[r] * inv;
    orow[48 + l16] = o3[r] * inv;
  }
}

// ---------------------------------------------------------------------------
// Kernel 3: output projection (ROWS x 1024) @ Wo(1024 x 1024) -> d_out.
// Each wave computes a 32x64 strip (B-fragments reused across 2 M-tiles,
// A-fragments across 4 N-tiles; all fragment loads contiguous b64).
// ---------------------------------------------------------------------------
__global__ __launch_bounds__(256) void out_proj_kernel(
    const float* __restrict__ A, const float* __restrict__ Wot,
    float* __restrict__ C) {
  const int wave = threadIdx.x >> 5;
  const int lane = threadIdx.x & 31;
  const int half = lane >> 4;
  const int l16  = lane & 15;

  const int row0 = blockIdx.x * 32;
  const int col0 = (blockIdx.y * 8 + wave) * 64;

  v8f c00 = {}, c01 = {}, c02 = {}, c03 = {};
  v8f c10 = {}, c11 = {}, c12 = {}, c13 = {};
  const float* xrow0 = A + (size_t)(row0 + l16) * D_MODEL;
  const float* xrow1 = xrow0 + (size_t)16 * D_MODEL;
  const float* w0 = Wot + (size_t)(col0 +  0 + l16) * D_MODEL;
  const float* w1 = Wot + (size_t)(col0 + 16 + l16) * D_MODEL;
  const float* w2 = Wot + (size_t)(col0 + 32 + l16) * D_MODEL;
  const float* w3 = Wot + (size_t)(col0 + 48 + l16) * D_MODEL;

#pragma unroll 4
  for (int k0 = 0; k0 < D_MODEL; k0 += 4) {
    const int kr = k0 + 2 * half;
    v2f a0 = *(const v2f*)(xrow0 + kr);
    v2f a1 = *(const v2f*)(xrow1 + kr);
    v2f b0 = *(const v2f*)(w0 + kr);
    v2f b1 = *(const v2f*)(w1 + kr);
    v2f b2 = *(const v2f*)(w2 + kr);
    v2f b3 = *(const v2f*)(w3 + kr);
    c00 = wmma_f32(a0, b0, c00);  c10 = wmma_f32(a1, b0, c10);
    c01 = wmma_f32(a0, b1, c01);  c11 = wmma_f32(a1, b1, c11);
    c02 = wmma_f32(a0, b2, c02);  c12 = wmma_f32(a1, b2, c12);
    c03 = wmma_f32(a0, b3, c03);  c13 = wmma_f32(a1, b3, c13);
  }

#pragma unroll
  for (int r = 0; r < 8; ++r) {
    const int row = row0 + r + 8 * half;
    float* crow0 = C + (size_t)row * D_MODEL + col0;
    float* crow1 = crow0 + (size_t)16 * D_MODEL;
    crow0[ 0 + l16] = c00[r];
    crow0[16 + l16] = c01[r];
    crow0[32 + l16] = c02[r];
    crow0[48 + l16] = c03[r];
    crow1[ 0 + l16] = c10[r];
    crow1[16 + l16] = c11[r];
    crow1[32 + l16] = c12[r];
    crow1[48 + l16] = c13[r];
  }
}

extern "C" void kernel_launch(void* const* d_in, const int* in_sizes, int n_in,
                              void* d_out, int out_size, void* d_ws, size_t ws_size,
                              hipStream_t stream) {
  const float* x  = (const float*)d_in[0];
  const float* Wq = (const float*)d_in[1];
  const float* Wk = (const float*)d_in[2];
  const float* Wv = (const float*)d_in[3];
  const float* Wo = (const float*)d_in[4];

  float* ws = (float*)d_ws;
  const size_t per = (size_t)BATCH * N_HEADS * SEQ * D_HEAD;  // 8,388,608 (32 MB)
  const size_t wsz = (size_t)D_MODEL * D_MODEL;               // 1,048,576 (4 MB)
  float* Q   = ws;
  float* K   = ws + per;
  float* Vt  = ws + 2 * per;
  float* AO  = ws + 3 * per;          // attention output, (B, S, H*Dh)
  float* Wqt = ws + 4 * per;
  float* Wkt = Wqt + wsz;
  float* Wvt = Wkt + wsz;
  float* Wot = Wvt + wsz;

  dim3 g0(D_MODEL / 32, D_MODEL / 32, 4);             // (32, 32, 4)
  transpose_w_kernel<<<g0, 256, 0, stream>>>(Wq, Wk, Wv, Wo, Wqt, Wkt, Wvt, Wot);

  dim3 g1(ROWS / 32, (N_HEADS * D_HEAD) / (16 * 8));  // (256, 8)
  qkv_proj_kernel<<<g1, 256, 0, stream>>>(x, Wqt, Wkt, Wvt, Q, K, Vt);

  dim3 g2(SEQ / (16 * 8), BATCH * N_HEADS);           // (16, 64)
  flash_attn_kernel<<<g2, 256, 0, stream>>>(Q, K, Vt, AO);

  dim3 g3(ROWS / 32, D_MODEL / (64 * 8));             // (256, 2)
  out_proj_kernel<<<g3, 256, 0, stream>>>(AO, Wot, (float*)d_out);
}